// MLP_17214228922441
// MI455X (gfx1250) — compile-verified
//
#include <hip/hip_runtime.h>

typedef __attribute__((ext_vector_type(2))) float v2f;
typedef __attribute__((ext_vector_type(8))) float v8f;

#define K_DIM         784
#define ROWS_PER_WG   128
#define THREADS       256
#define K_TILE        56            // 784 = 14 * 56
#define K_CHUNKS      14
#define LDS_STRIDE    60            // K_TILE + 4 pad -> conflict-free A-fragment reads
#define F4_PER_ROW    (K_TILE / 4)  // 14
#define F4_PER_CHUNK  (ROWS_PER_WG * F4_PER_ROW)  // 1792
#define F4_PER_THREAD (F4_PER_CHUNK / THREADS)    // 7

// ---- CDNA5 async global -> LDS copy (16B per lane), tracked by ASYNCcnt ----
__device__ __forceinline__ void async_cp16(const float* gp, float* lp)
{
    // Low 32 bits of a flat LDS address == LDS byte offset (ISA 10.2 aperture rule).
    unsigned loff = (unsigned)(unsigned long long)lp;
    asm volatile("global_load_async_to_lds_b128 %0, %1, off"
                 :: "v"(loff), "v"(gp)
                 : "memory");
}

__device__ __forceinline__ void issue_chunk(const float* __restrict__ x,
                                            int rowBase, int k0,
                                            float* __restrict__ dst, int t)
{
#pragma unroll
    for (int i = 0; i < F4_PER_THREAD; ++i) {
        int q   = i * THREADS + t;          // float4 index within tile
        int row = q / F4_PER_ROW;
        int c4  = q - row * F4_PER_ROW;
        const float* gp = x + (size_t)(rowBase + row) * K_DIM + (k0 + c4 * 4);
        float*       lp = dst + row * LDS_STRIDE + c4 * 4;
        async_cp16(gp, lp);
    }
}

__device__ __forceinline__ void compute_chunk(const float* __restrict__ w, int k0,
                                              const float* __restrict__ buf,
                                              int waveRow, int mrow, int khalf,
                                              v8f& acc0, v8f& acc1)
{
    const float* wp = w + k0 + khalf;
    const float* lp = buf + (waveRow + mrow) * LDS_STRIDE + khalf;
#pragma unroll
    for (int s = 0; s < K_TILE / 4; s += 2) {
        v2f a0 = *(const v2f*)(lp + s * 4);
        v2f b0 = *(const v2f*)(wp + s * 4);
        acc0 = __builtin_amdgcn_wmma_f32_16x16x4_f32(
            false, a0, false, b0, (short)0, acc0, false, false);
        v2f a1 = *(const v2f*)(lp + s * 4 + 4);
        v2f b1 = *(const v2f*)(wp + s * 4 + 4);
        acc1 = __builtin_amdgcn_wmma_f32_16x16x4_f32(
            false, a1, false, b1, (short)0, acc1, false, false);
    }
}

__global__ __launch_bounds__(THREADS)
void matvec_wmma_async(const float* __restrict__ x,
                       const float* __restrict__ w,
                       const float* __restrict__ bias_p,
                       float* __restrict__ out)
{
    __shared__ float buf[2][ROWS_PER_WG * LDS_STRIDE];  // 2 x 30720 B = 61440 B

    const int t       = threadIdx.x;
    const int wave    = t >> 5;
    const int lane    = t & 31;
    const int rowBase = blockIdx.x * ROWS_PER_WG;

    // WMMA 16x4 f32 A-layout: lane L(<16): row M=L, K={0,1}; lane L+16: row M=L, K={2,3}
    const int mrow    = lane & 15;
    const int khalf   = (lane >> 4) * 2;
    const int waveRow = wave * 16;

    v8f acc0 = {};
    v8f acc1 = {};

    // prologue: start chunk 0 into buf[0]
    issue_chunk(x, rowBase, 0, buf[0], t);

    // steady state: overlap async fill of buf[(kc+1)&1] with WMMA on buf[kc&1]
    for (int kc = 0; kc < K_CHUNKS - 1; ++kc) {
        issue_chunk(x, rowBase, (kc + 1) * K_TILE, buf[(kc + 1) & 1], t);
        // 7 newly issued outstanding; <=7 means chunk kc's 7 copies retired (in-order)
        asm volatile("s_wait_asynccnt 0x7" ::: "memory");
        __syncthreads();
        compute_chunk(w, kc * K_TILE, buf[kc & 1], waveRow, mrow, khalf, acc0, acc1);
        __syncthreads();   // all reads of buf[kc&1] done before it is refilled
    }

    // epilogue: last chunk
    asm volatile("s_wait_asynccnt 0x0" ::: "memory");
    __syncthreads();
    compute_chunk(w, (K_CHUNKS - 1) * K_TILE, buf[(K_CHUNKS - 1) & 1],
                  waveRow, mrow, khalf, acc0, acc1);

    // writeback: D columns identical; lane0 holds M=0..7, lane16 holds M=8..15
    const float bias = bias_p[0];
    if (lane == 0) {
#pragma unroll
        for (int m = 0; m < 8; ++m)
            out[rowBase + waveRow + m] = acc0[m] + acc1[m] + bias;
    } else if (lane == 16) {
#pragma unroll
        for (int m = 0; m < 8; ++m)
            out[rowBase + waveRow + 8 + m] = acc0[m] + acc1[m] + bias;
    }
}

extern "C" void kernel_launch(void* const* d_in, const int* in_sizes, int n_in,
                              void* d_out, int out_size, void* d_ws, size_t ws_size,
                              hipStream_t stream)
{
    const float* x = (const float*)d_in[0];
    const float* w = (const float*)d_in[1];
    const float* b = (const float*)d_in[2];
    float* out     = (float*)d_out;

    const int nRows = in_sizes[0] / K_DIM;      // 65536
    const int grid  = nRows / ROWS_PER_WG;      // 512

    matvec_wmma_async<<<grid, THREADS, 0, stream>>>(x, w, b, out);
}